// FALayer_1589137899749
// MI455X (gfx1250) — compile-verified
//
#include <hip/hip_runtime.h>
#include <hip/hip_bf16.h>

// FALayer (FAGCN-style) edge-gated aggregation for MI455X / gfx1250.
//
// Reference:  g_e = tanh(Wd.h[dst] + Ws.h[src] + b); e = g*deg[dst]*deg[src];
//             z[v] = sum_{(u->v)} h[u]*e
//
// Optimization: the gate is linear per endpoint, so precompute per-node
//   sdst[v] = Wd.h[v],  ssrc[v] = Ws.h[v]      (V_WMMA_F32_16X16X4_F32)
// then each edge needs only scalars + one 512B row gather + 128 f32 atomics.
// h (25.6MB) and z (25.6MB) are L2-resident (192MB L2), so the scatter runs
// at L2/atomic rate; HBM traffic is ~60MB total (~3us at 23.3 TB/s).

typedef __attribute__((ext_vector_type(2))) float v2f;
typedef __attribute__((ext_vector_type(8))) float v8f;

#define DFEAT 128

// ---------------------------------------------------------------------------
// Kernel A (dominant cost): edge scatter. One wave32 per edge; each lane owns
// 4 consecutive features (global_load_b128 gather of h[src], then 4x
// global_atomic_add_f32 into z[dst]). Per-edge scalars hit the same address
// across the wave (broadcast loads); tanh uses the f32 transcendental unit.
// Placed first in the file so its lowering shows in the disasm snippet
// (must be hardware f32 atomics, not a CAS loop).
// ---------------------------------------------------------------------------
__global__ void __launch_bounds__(256)
edge_scatter(const float* __restrict__ h,
             const float* __restrict__ deg,
             const float* __restrict__ gate_b,
             const float* __restrict__ sdst,
             const float* __restrict__ ssrc,
             const int*   __restrict__ src,
             const int*   __restrict__ dst,
             float* __restrict__ z,
             int n_edges) {
  const int e    = (int)(blockIdx.x * (blockDim.x >> 5) + (threadIdx.x >> 5));
  const int lane = (int)(threadIdx.x & 31u);
  if (e >= n_edges) return;

  const int s = src[e];
  const int d = dst[e];

  const float g  = tanhf(sdst[d] + ssrc[s] + gate_b[0]);
  const float ev = g * deg[d] * deg[s];

  const float4 hv = ((const float4*)(h + (size_t)s * DFEAT))[lane];
  float* zp = z + (size_t)d * DFEAT + (size_t)(lane << 2);

#if defined(__HIP_PLATFORM_AMD__) || defined(__AMDGCN__)
  unsafeAtomicAdd(zp + 0, hv.x * ev);   // -> global_atomic_add_f32
  unsafeAtomicAdd(zp + 1, hv.y * ev);
  unsafeAtomicAdd(zp + 2, hv.z * ev);
  unsafeAtomicAdd(zp + 3, hv.w * ev);
#else
  atomicAdd(zp + 0, hv.x * ev);
  atomicAdd(zp + 1, hv.y * ev);
  atomicAdd(zp + 2, hv.z * ev);
  atomicAdd(zp + 3, hv.w * ev);
#endif
}

// ---------------------------------------------------------------------------
// Kernel B: per-node gate dots via f32 WMMA.
// One wave32 per 16-node tile. A = H tile (16 x 4 f32), B = W tile (4 x 16),
// where only B columns 0 (Wd) and 1 (Ws) are nonzero (lane mask -> no
// divergence, EXEC all-1s as WMMA requires). K=128 in 32 WMMA steps, f32
// throughout to match the reference numerics.
//
// ISA f32 layouts (CDNA5 ISA 7.12.2), lane l: m = l&15, khalf = l>>4:
//   A vgpr j : element (M=m,            K = khalf*2 + j)
//   B vgpr j : element (K = khalf*2+j,  N = m)
//   C/D vgpr r: element (M = khalf*8+r, N = m)
// ---------------------------------------------------------------------------
__global__ void __launch_bounds__(256)
gate_dots_wmma(const float* __restrict__ h,
               const float* __restrict__ gate_w,   // [256]: [0:128]=Wd, [128:256]=Ws
               float* __restrict__ sdst,
               float* __restrict__ ssrc,
               int n_nodes, int n_tiles) {
  const int wave = (int)(blockIdx.x * (blockDim.x >> 5) + (threadIdx.x >> 5));
  const int lane = (int)(threadIdx.x & 31u);
  if (wave >= n_tiles) return;           // uniform per wave: EXEC stays all-1s

  const int m     = lane & 15;           // row (A) / col (B,C)
  const int khalf = lane >> 4;           // 0: K=0..1, 1: K=2..3 within a step

  int node = wave * 16 + m;
  if (node >= n_nodes) node = 0;         // safe clamp; result row is discarded
  const float* __restrict__ hrow = h + (size_t)node * DFEAT;

  // B column m: col0 = Wd, col1 = Ws, cols 2..15 = 0 (mask, no divergence)
  const float bmask = (m < 2) ? 1.0f : 0.0f;
  const int   boff  = (m == 1) ? DFEAT : 0;

  v8f c = {};
  #pragma unroll 4
  for (int k0 = 0; k0 < DFEAT; k0 += 4) {
    const int kb = k0 + khalf * 2;
    v2f a;
    a.x = hrow[kb];
    a.y = hrow[kb + 1];
    v2f b;
    b.x = gate_w[boff + kb]     * bmask;
    b.y = gate_w[boff + kb + 1] * bmask;
    // 8 args: (neg_a, A, neg_b, B, c_mod, C, reuse_a, reuse_b)
    c = __builtin_amdgcn_wmma_f32_16x16x4_f32(false, a, false, b,
                                              (short)0, c, false, false);
  }

  // Only columns 0 (sdst) and 1 (ssrc) carry real data.
  if (m < 2) {
    float* __restrict__ outp = ((m == 0) ? sdst : ssrc) + wave * 16 + khalf * 8;
    const int rowbase = wave * 16 + khalf * 8;
    if (rowbase + 8 <= n_nodes) {
      // Full tile (always true for N % 16 == 0): straight-line stores,
      // no per-row exec-mask juggling.
      #pragma unroll
      for (int r = 0; r < 8; ++r) outp[r] = c[r];
    } else {
      #pragma unroll
      for (int r = 0; r < 8; ++r)
        if (rowbase + r < n_nodes) outp[r] = c[r];
    }
  }
}

// ---------------------------------------------------------------------------
// Host-side launcher (graph-capture safe: only async, stream-ordered calls).
// Inputs (setup_inputs order): h[N*128] f32, deg[N] f32, gate_w[256] f32,
// gate_b[1] f32, src[E] i32, dst[E] i32.  Output: z[N*128] f32.
// ---------------------------------------------------------------------------
extern "C" void kernel_launch(void* const* d_in, const int* in_sizes, int n_in,
                              void* d_out, int out_size, void* d_ws, size_t ws_size,
                              hipStream_t stream) {
  const float* h      = (const float*)d_in[0];
  const float* deg    = (const float*)d_in[1];
  const float* gate_w = (const float*)d_in[2];
  const float* gate_b = (const float*)d_in[3];
  const int*   src    = (const int*)d_in[4];
  const int*   dst    = (const int*)d_in[5];
  float*       z      = (float*)d_out;

  const int N = in_sizes[1];   // deg has N elements
  const int E = in_sizes[4];   // src has E elements

  // Scratch: per-node gate partials (2 * N floats = 400 KB for N=50k).
  float* sdst = (float*)d_ws;
  float* ssrc = sdst + N;

  // z starts poisoned -> zero it (stream-ordered, capture-legal).
  hipMemsetAsync(d_out, 0, (size_t)out_size * sizeof(float), stream);

  // Per-node WMMA precompute: one wave per 16-node tile, 8 waves per block.
  const int n_tiles = (N + 15) / 16;                       // 3125 for N=50000
  const int blocks1 = (n_tiles + 7) / 8;
  gate_dots_wmma<<<blocks1, 256, 0, stream>>>(h, gate_w, sdst, ssrc, N, n_tiles);

  // Edge scatter: one wave per edge, 8 waves per block.
  const int blocks2 = (E + 7) / 8;                         // 100000 for E=800k
  edge_scatter<<<blocks2, 256, 0, stream>>>(h, deg, gate_b, sdst, ssrc,
                                            src, dst, z, E);
}